// CausalSelfAttention_2740189135366
// MI455X (gfx1250) — compile-verified
//
#include <hip/hip_runtime.h>
#include <hip/hip_bf16.h>
#include <math.h>

typedef __attribute__((ext_vector_type(2))) float v2f;
typedef __attribute__((ext_vector_type(8))) float v8f;

#define NB 2
#define NS 2048
#define NE 1024
#define NH 16
#define ND 64
#define NM (NB * NS)

static __device__ __forceinline__ v8f wmma_f32(v2f a, v2f b, v8f c) {
  // D = A(16x4 f32) x B(4x16 f32) + C(16x16 f32)
  return __builtin_amdgcn_wmma_f32_16x16x4_f32(false, a, false, b, (short)0, c,
                                               false, false);
}

// ---------------------------------------------------------------------------
// GEMM: dst = X[M x 1024] * W[1024 x 1024] + bias
// headsLayout=1 -> dst[b][h][s][d] (B,H,S,D); headsLayout=0 -> dst[row][col]
// Block tile 128x128, 8 waves (2x4), wave tile 32x64, K-block 16.
// W tile stored TRANSPOSED in LDS so B-fragments are contiguous v2f.
// ---------------------------------------------------------------------------
__global__ __launch_bounds__(256) void gemm128(const float* __restrict__ X,
                                               const float* __restrict__ W,
                                               const float* __restrict__ bias,
                                               float* __restrict__ dst,
                                               int headsLayout) {
  __shared__ float Xs[128][20];   // [row][k]  pad 20 (8B-aligned, no conflicts)
  __shared__ float Wf[128][20];   // [col][k]  transposed weight tile

  const int tid = threadIdx.x;
  const int wave = tid >> 5, lane = tid & 31;
  const int lm = lane & 15;
  const int half = lane >> 4;
  const int kbase = half << 1;
  const int wm = wave >> 1;       // 0..3 -> 32-row strip
  const int wn = wave & 1;        // 0..1 -> 64-col strip
  const int rowBase = blockIdx.y * 128;
  const int colBase = blockIdx.x * 128;

  v8f acc[2][4];
  for (int mi = 0; mi < 2; ++mi)
    for (int ni = 0; ni < 4; ++ni)
      for (int g = 0; g < 8; ++g) acc[mi][ni][g] = 0.0f;

  for (int kb = 0; kb < NE; kb += 16) {
    __syncthreads();
    // stage X tile: 128 rows x 16 k, coalesced float4
    for (int i = tid; i < 512; i += 256) {
      int r = i >> 2, c4 = (i & 3) << 2;
      *(float4*)&Xs[r][c4] =
          *(const float4*)&X[(size_t)(rowBase + r) * NE + kb + c4];
    }
    // stage W tile transposed: lanes vary k -> conflict-free LDS stores
    for (int i = tid; i < 512; i += 256) {
      int r = i & 15;             // k row
      int c4 = (i >> 4) << 2;     // col group
      float4 w = *(const float4*)&W[(size_t)(kb + r) * NE + colBase + c4];
      Wf[c4 + 0][r] = w.x;
      Wf[c4 + 1][r] = w.y;
      Wf[c4 + 2][r] = w.z;
      Wf[c4 + 3][r] = w.w;
    }
    __syncthreads();

#pragma unroll
    for (int t = 0; t < 4; ++t) {
      const int kk = (t << 2) + kbase;
      v2f a[2], b[4];
#pragma unroll
      for (int mi = 0; mi < 2; ++mi)
        a[mi] = *(const v2f*)&Xs[wm * 32 + mi * 16 + lm][kk];
#pragma unroll
      for (int ni = 0; ni < 4; ++ni)
        b[ni] = *(const v2f*)&Wf[wn * 64 + ni * 16 + lm][kk];
#pragma unroll
      for (int mi = 0; mi < 2; ++mi)
#pragma unroll
        for (int ni = 0; ni < 4; ++ni)
          acc[mi][ni] = wmma_f32(a[mi], b[ni], acc[mi][ni]);
    }
  }

  // epilogue: bias + store
#pragma unroll
  for (int mi = 0; mi < 2; ++mi) {
#pragma unroll
    for (int ni = 0; ni < 4; ++ni) {
      const int col = colBase + wn * 64 + ni * 16 + lm;
      const float bv = bias[col];
#pragma unroll
      for (int g = 0; g < 8; ++g) {
        const int row = rowBase + wm * 32 + mi * 16 + g + half * 8;
        const float val = acc[mi][ni][g] + bv;
        if (headsLayout) {
          const int b = row >> 11;           // row / S
          const int s = row & (NS - 1);
          const int h = col >> 6;            // col / D
          const int d = col & (ND - 1);
          dst[(((size_t)b * NH + h) * NS + s) * ND + d] = val;
        } else {
          dst[(size_t)row * NE + col] = val;
        }
      }
    }
  }
}

// ---------------------------------------------------------------------------
// Causal flash attention: one (b,h) per blockIdx.y, 128 query rows per block,
// 16 query rows per wave. K staged row-major, V staged transposed (d-major)
// so all B-fragments are contiguous v2f.
// ---------------------------------------------------------------------------
__global__ __launch_bounds__(256) void attn128(const float* __restrict__ Q,
                                               const float* __restrict__ K,
                                               const float* __restrict__ V,
                                               float* __restrict__ out) {
  __shared__ float Ks[64][68];      // [key][d]
  __shared__ float Vt[64][68];      // [d][key]  transposed
  __shared__ float Pb[8][16][18];   // per-wave P transpose buffer

  const int tid = threadIdx.x;
  const int wave = tid >> 5, lane = tid & 31;
  const int lm = lane & 15;
  const int half = lane >> 4;
  const int kbase = half << 1;
  const int bh = blockIdx.y;
  const int q0 = blockIdx.x * 128;
  const int qr0 = q0 + wave * 16;   // first query row of this wave

  const float* Qp = Q + (size_t)bh * NS * ND;
  const float* Kp = K + (size_t)bh * NS * ND;
  const float* Vp = V + (size_t)bh * NS * ND;

  // Q tile (16 rows x 64) as 16 A-fragments, kept in registers
  v2f qf[16];
#pragma unroll
  for (int t = 0; t < 16; ++t)
    qf[t] = *(const v2f*)&Qp[(size_t)(qr0 + lm) * ND + t * 4 + kbase];

  v8f o[4];
  float mst[8], lst[8];
  for (int nf = 0; nf < 4; ++nf)
    for (int g = 0; g < 8; ++g) o[nf][g] = 0.0f;
  for (int g = 0; g < 8; ++g) { mst[g] = -INFINITY; lst[g] = 0.0f; }

  const int ktiles = (q0 >> 6) + 2;   // keys needed: [0, q0+128)
  for (int kt = 0; kt < ktiles; ++kt) {
    const int kb0 = kt << 6;
    __syncthreads();
    // K tile: row-major, coalesced
    for (int i = tid; i < 1024; i += 256) {
      int r = i >> 4, c4 = (i & 15) << 2;
      *(float4*)&Ks[r][c4] = *(const float4*)&Kp[(size_t)(kb0 + r) * ND + c4];
    }
    // V tile: transposed; lanes vary key -> conflict-free LDS stores
    for (int i = tid; i < 1024; i += 256) {
      int r = i & 63;             // key
      int c4 = (i >> 6) << 2;     // d group
      float4 w = *(const float4*)&Vp[(size_t)(kb0 + r) * ND + c4];
      Vt[c4 + 0][r] = w.x;
      Vt[c4 + 1][r] = w.y;
      Vt[c4 + 2][r] = w.z;
      Vt[c4 + 3][r] = w.w;
    }
    __syncthreads();

    for (int sub = 0; sub < 4; ++sub) {
      const int kkb = kb0 + sub * 16;
      if (kkb > qr0 + 15) break;      // fully above diagonal for this wave

      // scores = Q * K^T  (16q x 16k, K-dim 64)
      v8f c;
      for (int g = 0; g < 8; ++g) c[g] = 0.0f;
#pragma unroll
      for (int t = 0; t < 16; ++t) {
        v2f bf = *(const v2f*)&Ks[sub * 16 + lm][t * 4 + kbase];
        c = wmma_f32(qf[t], bf, c);
      }

      // online softmax (rows live in 16-lane groups; col = lane%16)
      const int kcol = kkb + lm;
      float pr[8], al[8];
#pragma unroll
      for (int g = 0; g < 8; ++g) {
        const int qq = qr0 + g + half * 8;
        float s = c[g] * 0.125f;                 // 1/sqrt(64)
        s = (kcol <= qq) ? s : -INFINITY;
        float mx = s;
        mx = fmaxf(mx, __shfl_xor(mx, 1, 32));
        mx = fmaxf(mx, __shfl_xor(mx, 2, 32));
        mx = fmaxf(mx, __shfl_xor(mx, 4, 32));
        mx = fmaxf(mx, __shfl_xor(mx, 8, 32));
        const float mnew = fmaxf(mst[g], mx);
        const float a = __expf(mst[g] - mnew);
        const float p = __expf(s - mnew);
        float rs = p;
        rs += __shfl_xor(rs, 1, 32);
        rs += __shfl_xor(rs, 2, 32);
        rs += __shfl_xor(rs, 4, 32);
        rs += __shfl_xor(rs, 8, 32);
        lst[g] = lst[g] * a + rs;
        mst[g] = mnew;
        al[g] = a;
        pr[g] = p;
      }

      // P: C-layout -> A-layout via wave-private LDS; rescale O
#pragma unroll
      for (int g = 0; g < 8; ++g) Pb[wave][g + half * 8][lm] = pr[g];
#pragma unroll
      for (int nf = 0; nf < 4; ++nf)
#pragma unroll
        for (int g = 0; g < 8; ++g) o[nf][g] *= al[g];

      asm volatile("s_wait_dscnt 0" ::: "memory");

      v2f pf[4];
#pragma unroll
      for (int t = 0; t < 4; ++t)
        pf[t] = *(const v2f*)&Pb[wave][lm][t * 4 + kbase];

      // O += P * V   (16q x 64d, K-dim 16)
#pragma unroll
      for (int nf = 0; nf < 4; ++nf) {
        v8f oc = o[nf];
#pragma unroll
        for (int t = 0; t < 4; ++t) {
          v2f bf = *(const v2f*)&Vt[nf * 16 + lm][sub * 16 + t * 4 + kbase];
          oc = wmma_f32(pf[t], bf, oc);
        }
        o[nf] = oc;
      }
    }
  }

  // normalize + store to [B,S,E] (pre-output-projection)
  const int b = bh >> 4;    // bh / H
  const int h = bh & 15;    // bh % H
#pragma unroll
  for (int g = 0; g < 8; ++g) {
    const float inv = 1.0f / lst[g];
    const int s = qr0 + g + half * 8;
#pragma unroll
    for (int nf = 0; nf < 4; ++nf) {
      const int d = nf * 16 + lm;
      out[((size_t)b * NS + s) * NE + h * ND + d] = o[nf][g] * inv;
    }
  }
}

// ---------------------------------------------------------------------------
extern "C" void kernel_launch(void* const* d_in, const int* in_sizes, int n_in,
                              void* d_out, int out_size, void* d_ws,
                              size_t ws_size, hipStream_t stream) {
  const float* x  = (const float*)d_in[0];
  const float* Wq = (const float*)d_in[1];
  const float* bq = (const float*)d_in[2];
  const float* Wk = (const float*)d_in[3];
  const float* bk = (const float*)d_in[4];
  const float* Wv = (const float*)d_in[5];
  const float* bv = (const float*)d_in[6];
  const float* Wo = (const float*)d_in[7];
  const float* bo = (const float*)d_in[8];
  float* out = (float*)d_out;

  const size_t qkvElems = (size_t)NB * NH * NS * ND;   // 4 Mi floats each
  float* qb = (float*)d_ws;
  float* kb = qb + qkvElems;
  float* vb = kb + qkvElems;
  float* ab = vb + qkvElems;   // attention output [B,S,E]

  dim3 gg(NE / 128, NM / 128);          // (8, 32)
  gemm128<<<gg, 256, 0, stream>>>(x, Wq, bq, qb, 1);
  gemm128<<<gg, 256, 0, stream>>>(x, Wk, bk, kb, 1);
  gemm128<<<gg, 256, 0, stream>>>(x, Wv, bv, vb, 1);

  attn128<<<dim3(NS / 128, NB * NH), 256, 0, stream>>>(qb, kb, vb, ab);

  gemm128<<<gg, 256, 0, stream>>>(ab, Wo, bo, out, 0);
}